// FasterGRUTree_35450660061655
// MI455X (gfx1250) — compile-verified
//
#include <hip/hip_runtime.h>

// FasterGRUTree on MI455X (gfx1250), fp32 end-to-end via V_WMMA_F32_16X16X4_F32.
//
// Layout assumptions (CDNA5 ISA 7.12.2, wave32):
//   A (16x4 f32): lane m (0..15) v0=A[m][k],v1=A[m][k+1]; lanes 16..31 hold k+2,k+3
//   B (4x16 f32): lane n (0..15) v0=B[k][n],v1=B[k+1][n]; lanes 16..31 hold k+2,k+3
//   C/D (16x16 f32): vgpr i, lanes 0..15 -> D[i][n]; lanes 16..31 -> D[i+8][n]

typedef float v2f __attribute__((ext_vector_type(2)));
typedef float v8f __attribute__((ext_vector_type(8)));

#define MEM   512
#define TAG   100
#define WORD  300
#define KCH   (MEM + 2 * TAG)   // 712, children-GRU input width
#define KLEAF (WORD + TAG)      // 400, leaf-GRU input width
#define HPAD  516               // 512 + 4 float pad -> conflict-free LDS

__device__ __forceinline__ float sigmoidf_(float x) {
    return 1.0f / (1.0f + __expf(-x));
}

__device__ __forceinline__ v8f wmma4(v2f a, v2f b, v8f c) {
    return __builtin_amdgcn_wmma_f32_16x16x4_f32(
        /*neg_a=*/false, a, /*neg_b=*/false, b,
        /*c_mod=*/(short)0, c, /*reuse_a=*/false, /*reuse_b=*/false);
}

// Accumulate `steps` K=4 WMMA steps: one shared A stream, three B (gate) streams.
__device__ __forceinline__ void mm_region3(const float* __restrict__ pa,
                                           const float* __restrict__ pr,
                                           const float* __restrict__ pz,
                                           const float* __restrict__ pn,
                                           int steps, v8f& aR, v8f& aZ, v8f& aN) {
    for (int k = 0; k < steps; ++k) {
        v2f a  = *(const v2f*)pa; pa += 4;
        v2f br = *(const v2f*)pr; pr += 4;
        v2f bz = *(const v2f*)pz; pz += 4;
        v2f bn = *(const v2f*)pn; pn += 4;
        aR = wmma4(a, br, aR);
        aZ = wmma4(a, bz, aZ);
        aN = wmma4(a, bn, aN);
    }
}

// ---------------------------------------------------------------------------
// Leaf pass: states[L-1 + i] = GRU(cat(embs[i], tags[L-1+i]), h=0)
// One block = 16 rows; 8 waves x 4 column-groups cover all 512 hidden units.
// ---------------------------------------------------------------------------
__global__ __launch_bounds__(256) void gru_leaf_kernel(
    const float* __restrict__ embs, const float* __restrict__ tags,
    const float* __restrict__ Wi, const float* __restrict__ bi,
    const float* __restrict__ bh, float* __restrict__ states, int L) {
    const int tid  = threadIdx.x;
    const int wave = tid >> 5;
    const int lane = tid & 31;
    const int ln   = lane & 15;
    const int half = lane >> 4;
    const int kh   = half * 2;

    const int rowBase = blockIdx.x * 16;
    const int li      = min(rowBase + ln, L - 1);      // leaf index for A gather
    const int node    = (L - 1) + li;                  // heap id of this leaf

    for (int gi = 0; gi < 4; ++gi) {
        const int col = (wave + 8 * gi) * 16 + ln;
        const float* pWr = Wi + (size_t)(col          ) * KLEAF + kh;
        const float* pWz = Wi + (size_t)(col +     MEM) * KLEAF + kh;
        const float* pWn = Wi + (size_t)(col + 2 * MEM) * KLEAF + kh;

        v8f aR = {}; v8f aZ = {}; v8f aN = {};
        mm_region3(embs + (size_t)li * WORD + kh, pWr, pWz, pWn, WORD / 4, aR, aZ, aN);
        pWr += WORD; pWz += WORD; pWn += WORD;
        mm_region3(tags + (size_t)node * TAG + kh, pWr, pWz, pWn, TAG / 4, aR, aZ, aN);

        const float bir = bi[col], biz = bi[col + MEM], bin = bi[col + 2 * MEM];
        const float bhr = bh[col], bhz = bh[col + MEM], bhn = bh[col + 2 * MEM];
#pragma unroll
        for (int i = 0; i < 8; ++i) {
            const int rowg = rowBase + i + 8 * half;
            float r  = sigmoidf_(aR[i] + bir + bhr);
            float z  = sigmoidf_(aZ[i] + biz + bhz);
            float nn = tanhf(aN[i] + bin + r * bhn);
            if (rowg < L)
                states[((size_t)(L - 1) + rowg) * MEM + col] = (1.0f - z) * nn;
        }
    }
}

// ---------------------------------------------------------------------------
// One internal tree level (n = 2^lvl rows). Three fused GRU stages per block:
//   h1 = GRU(xl, 0); h2 = GRU(xr, h1); st = GRU(0-width-1, h2)
// h1/h2 shared across waves via one 33KB LDS buffer; this wave's h2 tiles
// additionally stay in registers for the z*h2 term of stage 3.
// ---------------------------------------------------------------------------
__global__ __launch_bounds__(256) void gru_level_kernel(
    float* __restrict__ states, const float* __restrict__ tags,
    const float* __restrict__ chWi, const float* __restrict__ chWh,
    const float* __restrict__ chBi, const float* __restrict__ chBh,
    const float* __restrict__ ndWh, const float* __restrict__ ndBi,
    const float* __restrict__ ndBh, int nLevel) {
    __shared__ float hbuf[16 * HPAD];

    const int tid  = threadIdx.x;
    const int wave = tid >> 5;
    const int lane = tid & 31;
    const int ln   = lane & 15;
    const int half = lane >> 4;
    const int kh   = half * 2;

    const int rowBase = blockIdx.x * 16;
    const int rg = min(rowBase + ln, nLevel - 1);   // clamped row for gathers
    const int p  = (nLevel - 1) + rg;               // parent heap id
    const int lc = 2 * p + 1;
    const int rc = 2 * p + 2;

    float h2reg[4][8];

    // ---- Stage 1: h1 = GRU(xl, 0), xl = cat(states[lc], tags[lc], tags[p])
    for (int gi = 0; gi < 4; ++gi) {
        const int col = (wave + 8 * gi) * 16 + ln;
        const float* pWr = chWi + (size_t)(col          ) * KCH + kh;
        const float* pWz = chWi + (size_t)(col +     MEM) * KCH + kh;
        const float* pWn = chWi + (size_t)(col + 2 * MEM) * KCH + kh;

        v8f aR = {}; v8f aZ = {}; v8f aN = {};
        mm_region3(states + (size_t)lc * MEM + kh, pWr, pWz, pWn, MEM / 4, aR, aZ, aN);
        pWr += MEM; pWz += MEM; pWn += MEM;
        mm_region3(tags + (size_t)lc * TAG + kh, pWr, pWz, pWn, TAG / 4, aR, aZ, aN);
        pWr += TAG; pWz += TAG; pWn += TAG;
        mm_region3(tags + (size_t)p * TAG + kh, pWr, pWz, pWn, TAG / 4, aR, aZ, aN);

        const float bir = chBi[col], biz = chBi[col + MEM], bin = chBi[col + 2 * MEM];
        const float bhr = chBh[col], bhz = chBh[col + MEM], bhn = chBh[col + 2 * MEM];
#pragma unroll
        for (int i = 0; i < 8; ++i) {
            const int rowi = i + 8 * half;
            float r  = sigmoidf_(aR[i] + bir + bhr);
            float z  = sigmoidf_(aZ[i] + biz + bhz);
            float nn = tanhf(aN[i] + bin + r * bhn);
            hbuf[rowi * HPAD + col] = (1.0f - z) * nn;   // h1 (z*h term is 0)
        }
    }
    __syncthreads();

    // ---- Stage 2: h2 = GRU(xr, h1)
    for (int gi = 0; gi < 4; ++gi) {
        const int col = (wave + 8 * gi) * 16 + ln;
        const float* pWr = chWi + (size_t)(col          ) * KCH + kh;
        const float* pWz = chWi + (size_t)(col +     MEM) * KCH + kh;
        const float* pWn = chWi + (size_t)(col + 2 * MEM) * KCH + kh;

        v8f aR = {}; v8f aZ = {}; v8f aN = {}; v8f aHN = {};
        mm_region3(states + (size_t)rc * MEM + kh, pWr, pWz, pWn, MEM / 4, aR, aZ, aN);
        pWr += MEM; pWz += MEM; pWn += MEM;
        mm_region3(tags + (size_t)rc * TAG + kh, pWr, pWz, pWn, TAG / 4, aR, aZ, aN);
        pWr += TAG; pWz += TAG; pWn += TAG;
        mm_region3(tags + (size_t)p * TAG + kh, pWr, pWz, pWn, TAG / 4, aR, aZ, aN);

        // h-part: r,z gates merge into aR/aZ; n-gate's h-part kept separate (aHN)
        const float* pVr = chWh + (size_t)(col          ) * MEM + kh;
        const float* pVz = chWh + (size_t)(col +     MEM) * MEM + kh;
        const float* pVn = chWh + (size_t)(col + 2 * MEM) * MEM + kh;
        mm_region3(&hbuf[ln * HPAD + kh], pVr, pVz, pVn, MEM / 4, aR, aZ, aHN);

        const float bir = chBi[col], biz = chBi[col + MEM], bin = chBi[col + 2 * MEM];
        const float bhr = chBh[col], bhz = chBh[col + MEM], bhn = chBh[col + 2 * MEM];
#pragma unroll
        for (int i = 0; i < 8; ++i) {
            const int rowi = i + 8 * half;
            float r   = sigmoidf_(aR[i] + bir + bhr);
            float z   = sigmoidf_(aZ[i] + biz + bhz);
            float nn  = tanhf(aN[i] + bin + r * (aHN[i] + bhn));
            float h1v = hbuf[rowi * HPAD + col];
            h2reg[gi][i] = (1.0f - z) * nn + z * h1v;
        }
    }
    __syncthreads();                       // everyone done reading h1
    for (int gi = 0; gi < 4; ++gi) {       // publish h2 into the same buffer
        const int col = (wave + 8 * gi) * 16 + ln;
#pragma unroll
        for (int i = 0; i < 8; ++i)
            hbuf[(i + 8 * half) * HPAD + col] = h2reg[gi][i];
    }
    __syncthreads();

    // ---- Stage 3: st = GRU(zeros(width=1), h2)  -> gi = node_bi only
    for (int gi = 0; gi < 4; ++gi) {
        const int col = (wave + 8 * gi) * 16 + ln;
        const float* pVr = ndWh + (size_t)(col          ) * MEM + kh;
        const float* pVz = ndWh + (size_t)(col +     MEM) * MEM + kh;
        const float* pVn = ndWh + (size_t)(col + 2 * MEM) * MEM + kh;

        v8f aR = {}; v8f aZ = {}; v8f aN = {};
        mm_region3(&hbuf[ln * HPAD + kh], pVr, pVz, pVn, MEM / 4, aR, aZ, aN);

        const float bir = ndBi[col], biz = ndBi[col + MEM], bin = ndBi[col + 2 * MEM];
        const float bhr = ndBh[col], bhz = ndBh[col + MEM], bhn = ndBh[col + 2 * MEM];
#pragma unroll
        for (int i = 0; i < 8; ++i) {
            const int rowg = rowBase + i + 8 * half;
            float r  = sigmoidf_(aR[i] + bir + bhr);
            float z  = sigmoidf_(aZ[i] + biz + bhz);
            float nn = tanhf(bin + r * (aN[i] + bhn));
            float st = (1.0f - z) * nn + z * h2reg[gi][i];
            if (rowg < nLevel)
                states[((size_t)(nLevel - 1) + rowg) * MEM + col] = st;
        }
    }
}

__global__ void copy_root_kernel(const float* __restrict__ states,
                                 float* __restrict__ out) {
    int t = threadIdx.x;
    if (t < MEM) out[t] = states[t];
}

extern "C" void kernel_launch(void* const* d_in, const int* in_sizes, int n_in,
                              void* d_out, int out_size, void* d_ws, size_t ws_size,
                              hipStream_t stream) {
    (void)in_sizes; (void)n_in; (void)out_size; (void)ws_size;
    const float* embs   = (const float*)d_in[0];
    const float* tags   = (const float*)d_in[1];
    const float* leafWi = (const float*)d_in[2];
    // d_in[3] = leaf_Wh: unused (h=0 for leaves -> gh = bh)
    const float* leafBi = (const float*)d_in[4];
    const float* leafBh = (const float*)d_in[5];
    // d_in[6] = node_Wi: unused (node input is zeros -> gi = node_bi)
    const float* ndWh   = (const float*)d_in[7];
    const float* ndBi   = (const float*)d_in[8];
    const float* ndBh   = (const float*)d_in[9];
    const float* chWi   = (const float*)d_in[10];
    const float* chWh   = (const float*)d_in[11];
    const float* chBi   = (const float*)d_in[12];
    const float* chBh   = (const float*)d_in[13];

    const int DEPTH = 14;
    const int L = 1 << DEPTH;                  // 16384 leaves
    float* states = (float*)d_ws;              // needs (2L-1)*512*4 = 64 MB

    gru_leaf_kernel<<<L / 16, 256, 0, stream>>>(embs, tags, leafWi, leafBi,
                                                leafBh, states, L);
    for (int lvl = DEPTH - 1; lvl >= 0; --lvl) {
        const int n = 1 << lvl;
        const int blocks = (n + 15) / 16;
        gru_level_kernel<<<blocks, 256, 0, stream>>>(states, tags, chWi, chWh,
                                                     chBi, chBh, ndWh, ndBi,
                                                     ndBh, n);
    }
    copy_root_kernel<<<1, MEM, 0, stream>>>(states, (float*)d_out);
}